// Memory_21818433864235
// MI455X (gfx1250) — compile-verified
//
#include <hip/hip_runtime.h>
#include <hip/hip_bf16.h>

typedef float v2f __attribute__((ext_vector_type(2)));
typedef float v4f __attribute__((ext_vector_type(4)));
typedef float v8f __attribute__((ext_vector_type(8)));

#define LROW 128          // L
#define KH   512          // H
#define LL   16384        // L*L
#define LDSPITCH 36       // padded LDS row pitch (floats): 8B aligned, bank-spread

// ---------------------------------------------------------------------------
// Gram + double-centering. One block (256 thr = 8 waves) per matrix g.
// src rows are (128 x 512) fp32, row stride 65536 floats per matrix.
// If gidx != nullptr, matrix g is gathered as src + gidx[g]*65536.
// Outputs: Cout[g] = centered Gram (128x128), vout[g] = ||C||_F.
// ---------------------------------------------------------------------------
__global__ __launch_bounds__(256) void gram_center_kernel(
    const float* __restrict__ src, const int* __restrict__ gidx,
    float* __restrict__ Cout, float* __restrict__ vout)
{
    __shared__ float ldsA[LROW * LDSPITCH];   // 18 KB K-slice
    __shared__ float rowsum[LROW];
    __shared__ float rowmean[LROW];
    __shared__ float totMean;
    __shared__ float sqAcc;

    const int g    = blockIdx.x;
    const int tid  = threadIdx.x;
    const int w    = tid >> 5;       // wave id 0..7 -> tile-row strip
    const int lane = tid & 31;
    const int half = lane >> 4;      // K-half within f32 WMMA frags
    const int mm   = lane & 15;      // M (A-frag) / N (B-frag) within tile

    long long srow = gidx ? (long long)gidx[g] : (long long)g;
    const float* A = src + srow * (long long)(LROW * KH);

    v8f acc[8];
#pragma unroll
    for (int j = 0; j < 8; ++j) acc[j] = v8f{};

    for (int ks = 0; ks < KH; ks += 32) {
        // cooperative load of 128x32 slice (coalesced b128 loads)
#pragma unroll
        for (int i = 0; i < 4; ++i) {
            int f   = tid + i * 256;     // float4 index 0..1023
            int row = f >> 3;
            int c4  = (f & 7) << 2;
            v4f v = *(const v4f*)(A + row * KH + ks + c4);
            *(v4f*)(&ldsA[row * LDSPITCH + c4]) = v;
        }
        __syncthreads();
#pragma unroll
        for (int k0 = 0; k0 < 32; k0 += 4) {
            // A-frag: rows of strip w; identical lane layout for the B-frag
            v2f a = *(const v2f*)(&ldsA[(w * 16 + mm) * LDSPITCH + k0 + 2 * half]);
#pragma unroll
            for (int j = 0; j < 8; ++j) {
                v2f b = *(const v2f*)(&ldsA[(j * 16 + mm) * LDSPITCH + k0 + 2 * half]);
                acc[j] = __builtin_amdgcn_wmma_f32_16x16x4_f32(
                    false, a, false, b, (short)0, acc[j], false, false);
            }
        }
        __syncthreads();
    }

    // ---- row sums (Gram is symmetric: rowmean == colmean) ----
#pragma unroll
    for (int r = 0; r < 8; ++r) {
        float s = 0.f;
#pragma unroll
        for (int j = 0; j < 8; ++j) s += acc[j][r];
        // reduce across the 16 lanes of this half (same global row M)
        for (int off = 1; off < 16; off <<= 1) s += __shfl_xor(s, off, 32);
        if (mm == 0) rowsum[w * 16 + half * 8 + r] = s;
    }
    __syncthreads();
    if (tid < LROW) rowmean[tid] = rowsum[tid] * (1.0f / LROW);
    if (tid == 0) {
        float t = 0.f;
        for (int i = 0; i < LROW; ++i) t += rowsum[i];
        totMean = t * (1.0f / (LROW * LROW));
        sqAcc = 0.f;
    }
    __syncthreads();

    // ---- center, write C, accumulate ||C||^2 ----
    float* Cg = Cout + (long long)g * LL;
    float sq = 0.f;
    const float tm = totMean;
#pragma unroll
    for (int j = 0; j < 8; ++j) {
        int N = j * 16 + mm;
        float rmN = rowmean[N];
#pragma unroll
        for (int r = 0; r < 8; ++r) {
            int M = w * 16 + half * 8 + r;
            float c = acc[j][r] - rowmean[M] - rmN + tm;
            sq += c * c;
            Cg[M * LROW + N] = c;
        }
    }
    for (int off = 1; off < 32; off <<= 1) sq += __shfl_xor(sq, off, 32);
    if (lane == 0) atomicAdd(&sqAcc, sq);
    __syncthreads();
    if (tid == 0) vout[g] = sqrtf(sqAcc);
}

// ---------------------------------------------------------------------------
// Cross-Gram GEMM: out[mt*16+M][nt*16+N] = sum_k Arows[m][k]*Brows[n][k],
// K = 16384 (flattened LxL). One wave per 16x16 tile.
// ---------------------------------------------------------------------------
__global__ __launch_bounds__(32) void cross_gram_kernel(
    const float* __restrict__ Arows, const float* __restrict__ Brows,
    float* __restrict__ out, int K, int ldOut)
{
    const int lane = threadIdx.x & 31;
    const int half = lane >> 4;
    const int mm   = lane & 15;
    const int nt = blockIdx.x, mt = blockIdx.y;

    const float* ar = Arows + (long long)(mt * 16 + mm) * K + 2 * half;
    const float* br = Brows + (long long)(nt * 16 + mm) * K + 2 * half;

    v8f acc = v8f{};
    for (int k0 = 0; k0 < K; k0 += 16) {
#pragma unroll
        for (int u = 0; u < 4; ++u) {
            v2f a = *(const v2f*)(ar + k0 + u * 4);
            v2f b = *(const v2f*)(br + k0 + u * 4);
            acc = __builtin_amdgcn_wmma_f32_16x16x4_f32(
                false, a, false, b, (short)0, acc, false, false);
        }
    }
#pragma unroll
    for (int r = 0; r < 8; ++r)
        out[(mt * 16 + half * 8 + r) * ldOut + nt * 16 + mm] = acc[r];
}

// ---------------------------------------------------------------------------
// Loss matrix + first-occurrence argmax per batch row.
// ---------------------------------------------------------------------------
__global__ __launch_bounds__(256) void loss_argmax_kernel(
    const float* __restrict__ hsic, const float* __restrict__ vx,
    const float* __restrict__ vy, float* __restrict__ outMat,
    float* __restrict__ outIdxF, int* __restrict__ idxOut)
{
    __shared__ float bv[256];
    __shared__ int   bi[256];
    const int b = blockIdx.x, t = threadIdx.x;
    const float vxb = vx[b];
    float bestV = -__builtin_inff();
    int   bestI = 0;
    for (int m = t; m < 512; m += 256) {
        float val = -logf(fabsf(hsic[b * 512 + m] / (vxb * vy[m])) + 1e-8f);
        outMat[b * 512 + m] = val;
        if (val > bestV) { bestV = val; bestI = m; }
    }
    bv[t] = bestV; bi[t] = bestI;
    __syncthreads();
    for (int s = 128; s > 0; s >>= 1) {
        if (t < s) {
            float v2 = bv[t + s]; int i2 = bi[t + s];
            if (v2 > bv[t] || (v2 == bv[t] && i2 < bi[t])) { bv[t] = v2; bi[t] = i2; }
        }
        __syncthreads();
    }
    if (t == 0) { idxOut[b] = bi[0]; outIdxF[b] = (float)bi[0]; }
}

// ---------------------------------------------------------------------------
// Final scalar: -log(mean(|hsic2/(vx*vs)|) + eps)
// ---------------------------------------------------------------------------
__global__ __launch_bounds__(256) void final_loss_kernel(
    const float* __restrict__ hsic2, const float* __restrict__ vx,
    const float* __restrict__ vs, float* __restrict__ out0)
{
    __shared__ float red[256];
    const int t = threadIdx.x;
    float s = 0.f;
    for (int i = t; i < 1024; i += 256) {
        int b = i >> 5, c = i & 31;
        s += fabsf(hsic2[i] / (vx[b] * vs[c]));
    }
    red[t] = s;
    __syncthreads();
    for (int st = 128; st > 0; st >>= 1) {
        if (t < st) red[t] += red[t + st];
        __syncthreads();
    }
    if (t == 0) out0[0] = -logf(red[0] * (1.0f / 1024.0f) + 1e-8f);
}

// ---------------------------------------------------------------------------
extern "C" void kernel_launch(void* const* d_in, const int* in_sizes, int n_in,
                              void* d_out, int out_size, void* d_ws, size_t ws_size,
                              hipStream_t stream) {
    const float* x  = (const float*)d_in[0];   // (32,128,512)
    const float* cw = (const float*)d_in[1];   // (512, 65536)
    float* out = (float*)d_out;                // [loss(1) | matrix(16384) | idx(32)]
    float* ws  = (float*)d_ws;

    float* Cx  = ws;                    // 32  * 16384
    float* Cy  = Cx + 32 * LL;          // 512 * 16384
    float* Cs  = Cy + 512 * LL;         // 32  * 16384
    float* vx  = Cs + 32 * LL;          // 32
    float* vy  = vx + 32;               // 512
    float* vs  = vy + 512;              // 32
    float* hs  = vs + 32;               // 32*512
    float* hs2 = hs + 32 * 512;         // 32*32
    int*   idx = (int*)(hs2 + 1024);    // 32

    // 1) Cx, vx
    gram_center_kernel<<<32, 256, 0, stream>>>(x, nullptr, Cx, vx);
    // 2) Cy, vy for all 512 centroids (dominant work, bandwidth-bound)
    gram_center_kernel<<<512, 256, 0, stream>>>(cw, nullptr, Cy, vy);
    // 3) hsic(32x512) = Cx . Cy^T over K=16384 (WMMA)
    cross_gram_kernel<<<dim3(32, 2), 32, 0, stream>>>(Cx, Cy, hs, LL, 512);
    // 4) loss matrix + argmax
    loss_argmax_kernel<<<32, 256, 0, stream>>>(hs, vx, vy, out + 1, out + 1 + LL, idx);
    // 5) Cs, vs from gathered centroids
    gram_center_kernel<<<32, 256, 0, stream>>>(cw, idx, Cs, vs);
    // 6) hsic2(32x32) = Cx . Cs^T (WMMA)
    cross_gram_kernel<<<dim3(2, 2), 32, 0, stream>>>(Cx, Cs, hs2, LL, 32);
    // 7) scalar loss
    final_loss_kernel<<<1, 256, 0, stream>>>(hs2, vx, vs, out);
}